// DynamicConnectivity_37495064494226
// MI455X (gfx1250) — compile-verified
//
#include <hip/hip_runtime.h>

#define PRE   2048
#define POST  2048
#define BATCH 64
#define HIST  50

typedef __attribute__((ext_vector_type(2))) float v2f;
typedef __attribute__((ext_vector_type(8))) float v8f;

// ---------------------------------------------------------------------------
// Kernel 1: column means of pre/post spikes + EMA activity traces.
// threads [0,2048) -> pre, [2048,4096) -> post. Coalesced column sweeps.
// ws layout: ws[0..2047] = pre_mean, ws[2048..4095] = post_mean.
// ---------------------------------------------------------------------------
__global__ void dc_means_kernel(const float* __restrict__ pre_spikes,
                                const float* __restrict__ post_spikes,
                                const float* __restrict__ pre_act,
                                const float* __restrict__ post_act,
                                float* __restrict__ ws_means,
                                float* __restrict__ out_pre_act,
                                float* __restrict__ out_post_act) {
  int t = blockIdx.x * blockDim.x + threadIdx.x;
  if (t < PRE) {
    float s = 0.0f;
    #pragma unroll 4
    for (int b = 0; b < BATCH; ++b) s += pre_spikes[b * PRE + t];
    float m = s * (1.0f / (float)BATCH);
    ws_means[t] = m;
    out_pre_act[t] = 0.99f * pre_act[t] + 0.01f * m;
  } else if (t < PRE + POST) {
    int j = t - PRE;
    float s = 0.0f;
    #pragma unroll 4
    for (int b = 0; b < BATCH; ++b) s += post_spikes[b * POST + j];
    float m = s * (1.0f / (float)BATCH);
    ws_means[PRE + j] = m;
    out_post_act[j] = 0.99f * post_act[j] + 0.01f * m;
  }
}

// ---------------------------------------------------------------------------
// Kernel 2: new_lifetimes = lifetimes + mask (float4 vectorized elementwise).
// ---------------------------------------------------------------------------
__global__ void dc_lifetimes_kernel(const float* __restrict__ lifetimes,
                                    const float* __restrict__ mask,
                                    float* __restrict__ out) {
  int t = blockIdx.x * blockDim.x + threadIdx.x;
  const float4* a = (const float4*)lifetimes;
  const float4* b = (const float4*)mask;
  float4 x = a[t];
  float4 y = b[t];
  float4 r;
  r.x = x.x + y.x; r.y = x.y + y.y; r.z = x.z + y.z; r.w = x.w + y.w;
  ((float4*)out)[t] = r;
}

// ---------------------------------------------------------------------------
// Kernel 3: fused history copy + idx-slice scatter + per-pair mean.
// One wave per (i,j) pair: lanes 0..24 move an aligned float2 (50 floats),
// substitute outer(pre_mean,post_mean)[i,j] at h==idx, wave-reduce the sum.
// Non-temporal: single-use 839MB stream must not pollute L2.
// ---------------------------------------------------------------------------
__global__ void __launch_bounds__(256)
dc_history_kernel(const float* __restrict__ hist,
                  const float* __restrict__ ws_means,
                  const int* __restrict__ corr_idx_p,
                  float* __restrict__ out_hist,
                  float* __restrict__ out_avg) {
  const int lane = threadIdx.x & 31;
  const long long p = (long long)((blockIdx.x * blockDim.x + threadIdx.x) >> 5); // pair id
  const int i = (int)(p >> 11);     // POST == 2048
  const int j = (int)(p & 2047);
  const float corr = ws_means[i] * ws_means[PRE + j];
  const int idx = (((*corr_idx_p) % HIST) + HIST) % HIST;
  const long long base = p * HIST;

  float s = 0.0f;
  if (lane < HIST / 2) {
    const v2f* src = (const v2f*)(hist + base + 2 * lane);
    v2f v = __builtin_nontemporal_load(src);
    if (2 * lane     == idx) v.x = corr;
    if (2 * lane + 1 == idx) v.y = corr;
    __builtin_nontemporal_store(v, (v2f*)(out_hist + base + 2 * lane));
    s = v.x + v.y;
  }
  #pragma unroll
  for (int off = 16; off > 0; off >>= 1)
    s += __shfl_xor(s, off, 32);
  if (lane == 0) out_avg[p] = s * (1.0f / (float)HIST);
}

// ---------------------------------------------------------------------------
// Kernel 4: synaptic_current = pre_spikes @ (weights * mask) via
// V_WMMA_F32_16X16X4_F32. One wave per 16x16 output tile, 512 K-steps.
// A 16x4 layout: lane L holds A[M=L&15][K=2*(L>>4)+{0,1}]  -> float2 load.
// B 4x16 layout: lane L holds B[K=2*(L>>4)+{0,1}][N=L&15]  -> 2 loads, mask fused.
// C/D: VGPR r, lanes 0-15 -> M=r, lanes 16-31 -> M=r+8; N = lane&15.
// ---------------------------------------------------------------------------
__global__ void __launch_bounds__(256)
dc_wmma_gemm_kernel(const float* __restrict__ A,   // pre_spikes [64 x 2048]
                    const float* __restrict__ W,   // weights    [2048 x 2048]
                    const float* __restrict__ Mk,  // mask       [2048 x 2048]
                    float* __restrict__ C) {       // out        [64 x 2048]
  const int lane = threadIdx.x & 31;
  const int wave = (blockIdx.x * blockDim.x + threadIdx.x) >> 5; // 0..511
  const int mt   = wave >> 7;    // 0..3   (M tiles of 16 over BATCH=64)
  const int nt   = wave & 127;   // 0..127 (N tiles of 16 over POST=2048)
  const int half = lane >> 4;    // 0 or 1 -> K sub-pair
  const int sub  = lane & 15;

  const int arow = mt * 16 + sub;    // A row for this lane
  const int bcol = nt * 16 + sub;    // B col for this lane

  v8f acc = {};
  #pragma unroll 4
  for (int k0 = 0; k0 < PRE; k0 += 4) {
    const int k = k0 + 2 * half;
    v2f a = *(const v2f*)(A + (long long)arow * PRE + k);
    v2f b;
    b.x = W[(long long)(k    ) * POST + bcol] * Mk[(long long)(k    ) * POST + bcol];
    b.y = W[(long long)(k + 1) * POST + bcol] * Mk[(long long)(k + 1) * POST + bcol];
    acc = __builtin_amdgcn_wmma_f32_16x16x4_f32(
        /*neg_a=*/false, a, /*neg_b=*/false, b,
        /*c_mod=*/(short)0, acc, /*reuse_a=*/false, /*reuse_b=*/false);
  }

  #pragma unroll
  for (int r = 0; r < 8; ++r) {
    const int row = mt * 16 + r + 8 * half;
    C[(long long)row * POST + (nt * 16 + sub)] = acc[r];
  }
}

// ---------------------------------------------------------------------------
extern "C" void kernel_launch(void* const* d_in, const int* in_sizes, int n_in,
                              void* d_out, int out_size, void* d_ws, size_t ws_size,
                              hipStream_t stream) {
  const float* pre_spikes  = (const float*)d_in[0];
  const float* post_spikes = (const float*)d_in[1];
  // d_in[2] current_time: unused (plasticity branch skipped in reference path)
  const float* weights     = (const float*)d_in[3];
  const float* mask        = (const float*)d_in[4];
  const float* pre_act     = (const float*)d_in[5];
  const float* post_act    = (const float*)d_in[6];
  const float* hist        = (const float*)d_in[7];
  const float* lifetimes   = (const float*)d_in[8];
  const int*   corr_idx    = (const int*)d_in[9];

  float* out       = (float*)d_out;
  float* out_cur   = out;                          // [64,2048]
  float* out_preA  = out + (size_t)131072;         // [2048]
  float* out_postA = out + (size_t)133120;         // [2048]
  float* out_hist  = out + (size_t)135168;         // [2048,2048,50]
  float* out_life  = out + (size_t)209850368;      // [2048,2048]
  float* out_avg   = out + (size_t)214044672;      // [2048,2048]

  float* ws_means = (float*)d_ws;                  // 4096 floats = 16 KB

  // means + EMA traces (must precede history kernel; same stream => ordered)
  dc_means_kernel<<<(PRE + POST) / 256, 256, 0, stream>>>(
      pre_spikes, post_spikes, pre_act, post_act, ws_means, out_preA, out_postA);

  // lifetimes (float4 over 2048*2048)
  dc_lifetimes_kernel<<<(PRE * POST / 4) / 256, 256, 0, stream>>>(
      lifetimes, mask, out_life);

  // WMMA GEMM: 512 tiles = 64 blocks * 8 waves
  dc_wmma_gemm_kernel<<<64, 256, 0, stream>>>(
      pre_spikes, weights, mask, out_cur);

  // history stream: one wave per pair -> 4,194,304 waves = 524,288 blocks
  dc_history_kernel<<<(PRE * POST) / 8, 256, 0, stream>>>(
      hist, ws_means, corr_idx, out_hist, out_avg);
}